// SAGERegression0_51170240364594
// MI455X (gfx1250) — compile-verified
//
#include <hip/hip_runtime.h>

// GraphSAGE 2-layer forward on MI455X (gfx1250, wave32).
// Scatter-mean is the bottleneck (~1.6 GB of edge traffic, L2-resident atomics);
// the GEMMs use V_WMMA_F32_16X16X4_F32 for exact fp32 math at negligible cost.

#define N_NODES_C 100000
#define N_EDGES_C 1600000
#define FEATS     64          // IN_FEATS == HID_FEATS == 64
#define OUTF      32

typedef __attribute__((ext_vector_type(2))) float v2f;
typedef __attribute__((ext_vector_type(8))) float v8f;

// ---------------------------------------------------------------- zero fill
__global__ void sage_zero_f32(float* __restrict__ p, long long n) {
  long long i = (long long)blockIdx.x * blockDim.x + threadIdx.x;
  if (i < n) p[i] = 0.0f;
}

// ---------------------------------------------------------------- in-degree
__global__ void sage_degree(const long long* __restrict__ edges,
                            float* __restrict__ cnt) {
  int e = blockIdx.x * blockDim.x + threadIdx.x;
  if (e < N_EDGES_C) {
    long long d = edges[(long long)N_EDGES_C + e];
    atomicAdd(&cnt[d], 1.0f);
  }
}

// ------------------------------------------------- scatter-add of messages
// one thread per (edge, feature); consecutive threads cover one 64-float row
// -> coalesced gather reads and contiguous atomic bursts into L2.
__global__ void sage_scatter(const float* __restrict__ feat,
                             const long long* __restrict__ edges,
                             float* __restrict__ agg) {
  long long idx = (long long)blockIdx.x * blockDim.x + threadIdx.x;
  if (idx >= (long long)N_EDGES_C * FEATS) return;
  int e = (int)(idx >> 6);
  int f = (int)(idx & 63);
  long long s = edges[e];
  long long d = edges[(long long)N_EDGES_C + e];
  atomicAdd(&agg[d * FEATS + f], feat[s * FEATS + f]);
}

// ------------------------------------------------- fused SAGEConv GEMM
// out[16-node tile] = (agg*inv_deg) @ Wl + self @ Wr + bias  (optional ReLU)
// One wave per 16-node tile, NT 16-wide output tiles, K = 2*64 via
// V_WMMA_F32_16X16X4_F32 (exact fp32).
template <int NT, bool RELU>
__global__ __launch_bounds__(256)
void sage_gemm(const float* __restrict__ agg,   // [N_NODES][64]
               const float* __restrict__ self,  // [N_NODES][64]
               const float* __restrict__ Wl,    // [64][NT*16]
               const float* __restrict__ Wr,    // [64][NT*16]
               const float* __restrict__ bias,  // [NT*16]
               const float* __restrict__ cnt,   // [N_NODES]
               float* __restrict__ out) {       // [N_NODES][NT*16]
  const int lane  = threadIdx.x & 31;
  const int wave  = (int)(((long long)blockIdx.x * blockDim.x + threadIdx.x) >> 5);
  const int tiles = N_NODES_C / 16;            // 6250, exact
  if (wave >= tiles) return;

  const int NOUT  = NT * 16;
  const int base  = wave * 16;
  const int mrow  = lane & 15;                 // A row / B-&-D column index
  const int khalf = (lane >> 4) * 2;           // K sub-offset within group of 4
  const int node  = base + mrow;
  const float inv = 1.0f / fmaxf(cnt[node], 1.0f);   // mean normalization

  v8f c[NT] = {};

  // ---- aggregated-neighbor half: A = agg * inv, B = Wl
  #pragma unroll 4
  for (int kk = 0; kk < 16; ++kk) {
    const int k = kk * 4 + khalf;              // even -> 8B aligned
    v2f a = *(const v2f*)(agg + (long long)node * FEATS + k);
    a.x *= inv; a.y *= inv;
    #pragma unroll
    for (int nt = 0; nt < NT; ++nt) {
      const int n = nt * 16 + mrow;
      v2f b;
      b.x = Wl[(k    ) * NOUT + n];
      b.y = Wl[(k + 1) * NOUT + n];
      c[nt] = __builtin_amdgcn_wmma_f32_16x16x4_f32(
          false, a, false, b, (short)0, c[nt], false, false);
    }
  }

  // ---- self half: A = self, B = Wr
  #pragma unroll 4
  for (int kk = 0; kk < 16; ++kk) {
    const int k = kk * 4 + khalf;
    v2f a = *(const v2f*)(self + (long long)node * FEATS + k);
    #pragma unroll
    for (int nt = 0; nt < NT; ++nt) {
      const int n = nt * 16 + mrow;
      v2f b;
      b.x = Wr[(k    ) * NOUT + n];
      b.y = Wr[(k + 1) * NOUT + n];
      c[nt] = __builtin_amdgcn_wmma_f32_16x16x4_f32(
          false, a, false, b, (short)0, c[nt], false, false);
    }
  }

  // ---- epilogue: bias (+ReLU) and store.  D layout: VGPR r -> M = r + (lane/16)*8
  const int mout = base + (lane >> 4) * 8;
  #pragma unroll
  for (int nt = 0; nt < NT; ++nt) {
    const int n  = nt * 16 + mrow;
    const float bv = bias[n];
    #pragma unroll
    for (int r = 0; r < 8; ++r) {
      float v = c[nt][r] + bv;
      if (RELU) v = fmaxf(v, 0.0f);
      out[(long long)(mout + r) * NOUT + n] = v;
    }
  }
}

// ---------------------------------------------------------------- driver
extern "C" void kernel_launch(void* const* d_in, const int* in_sizes, int n_in,
                              void* d_out, int out_size, void* d_ws, size_t ws_size,
                              hipStream_t stream) {
  const float*     x     = (const float*)d_in[0];
  const long long* edges = (const long long*)d_in[1];   // int64 [2][E]
  const float*     W1l   = (const float*)d_in[2];
  const float*     W1r   = (const float*)d_in[3];
  const float*     b1    = (const float*)d_in[4];
  const float*     W2l   = (const float*)d_in[5];
  const float*     W2r   = (const float*)d_in[6];
  const float*     b2    = (const float*)d_in[7];
  float*           out   = (float*)d_out;

  // workspace layout: [agg | cnt | h]   (agg+cnt zeroed together)
  float* agg = (float*)d_ws;                                    // 6,400,000 f32
  float* cnt = agg + (long long)N_NODES_C * FEATS;              //   100,000 f32
  float* h   = cnt + N_NODES_C;                                 // 6,400,000 f32

  const long long aggN   = (long long)N_NODES_C * FEATS;
  const long long edgeWk = (long long)N_EDGES_C * FEATS;
  const int T = 256;
  const int zBlocksAC  = (int)((aggN + N_NODES_C + T - 1) / T);
  const int zBlocksA   = (int)((aggN + T - 1) / T);
  const int degBlocks  = (N_EDGES_C + T - 1) / T;
  const int scatBlocks = (int)((edgeWk + T - 1) / T);
  const int gemmBlocks = (N_NODES_C / 16 * 32 + T - 1) / T;     // 1 wave / 16 nodes

  // ---- layer 1
  sage_zero_f32<<<zBlocksAC, T, 0, stream>>>(agg, aggN + N_NODES_C);
  sage_degree  <<<degBlocks, T, 0, stream>>>(edges, cnt);
  sage_scatter <<<scatBlocks, T, 0, stream>>>(x, edges, agg);
  sage_gemm<4, true><<<gemmBlocks, T, 0, stream>>>(agg, x, W1l, W1r, b1, cnt, h);

  // ---- layer 2
  sage_zero_f32<<<zBlocksA, T, 0, stream>>>(agg, aggN);
  sage_scatter <<<scatBlocks, T, 0, stream>>>(h, edges, agg);
  sage_gemm<2, false><<<gemmBlocks, T, 0, stream>>>(agg, h, W2l, W2r, b2, cnt, out);
}